// SGC_90752658964691
// MI455X (gfx1250) — compile-verified
//
#include <hip/hip_runtime.h>
#include <hip/hip_bf16.h>

// ---------------------------------------------------------------------------
// SGC: 2-hop GCN propagation (scatter/gather, L2-resident) + f32 WMMA GEMM.
//   N=50000, E=800000, IN=128, OUT=64, K=2
// ---------------------------------------------------------------------------

typedef __attribute__((ext_vector_type(2))) float v2f;
typedef __attribute__((ext_vector_type(8))) float v8f;

// ---- degree / norm ---------------------------------------------------------

__global__ void sgc_init_deg(float* __restrict__ deg, int n) {
    int i = blockIdx.x * blockDim.x + threadIdx.x;
    if (i < n) deg[i] = 1.0f;               // self-loop contributes 1
}

__global__ void sgc_count_deg(const int* __restrict__ col,
                              float* __restrict__ deg, int e) {
    int i = blockIdx.x * blockDim.x + threadIdx.x;
    if (i < e) atomicAdd(&deg[col[i]], 1.0f);
}

__global__ void sgc_rsqrt(float* __restrict__ deg, int n) {
    int i = blockIdx.x * blockDim.x + threadIdx.x;
    if (i < n) {
        float d = deg[i];                    // d >= 1 always (self-loop)
        deg[i] = rsqrtf(d);
    }
}

// ---- propagation -----------------------------------------------------------
// One wave32 per node, 4 channels per lane (128 channels): self-loop term.
__global__ void sgc_self(const float* __restrict__ src,
                         const float* __restrict__ dinv,
                         float* __restrict__ dst, int n) {
    long long tid = (long long)blockIdx.x * blockDim.x + threadIdx.x;
    int i = (int)(tid >> 5);
    if (i >= n) return;
    int c = (int)(tid & 31) * 4;
    float w = dinv[i] * dinv[i];
    const float4 v = *(const float4*)(src + (size_t)i * 128 + c);
    float4 o = make_float4(v.x * w, v.y * w, v.z * w, v.w * w);
    *(float4*)(dst + (size_t)i * 128 + c) = o;
}

// One wave32 per edge, 4 channels per lane: float4 gather + 4 f32 atomics.
__global__ void sgc_edge(const float* __restrict__ src,
                         const int* __restrict__ row,
                         const int* __restrict__ col,
                         const float* __restrict__ dinv,
                         float* __restrict__ dst, int e) {
    long long tid = (long long)blockIdx.x * blockDim.x + threadIdx.x;
    int ei = (int)(tid >> 5);
    if (ei >= e) return;
    int c  = (int)(tid & 31) * 4;
    int r  = row[ei];
    int cl = col[ei];
    float w = dinv[r] * dinv[cl];
    const float4 v = *(const float4*)(src + (size_t)r * 128 + c);
    float* d = dst + (size_t)cl * 128 + c;
    atomicAdd(d + 0, v.x * w);
    atomicAdd(d + 1, v.y * w);
    atomicAdd(d + 2, v.z * w);
    atomicAdd(d + 3, v.w * w);
}

// ---- final projection with f32 WMMA ---------------------------------------
// out[N,64] = X[N,128] @ W[64,128]^T + bias.
// One wave computes a 16x64 strip: a single A fragment per k-step feeds 4
// WMMAs (one per 16-col B tile) -> A is read once, 4 wmma per A load.
// K=128 in steps of 4 via V_WMMA_F32_16X16X4_F32 (f32 precision preserved).
__global__ __launch_bounds__(256) void sgc_gemm_wmma(
    const float* __restrict__ X,    // [n,128]
    const float* __restrict__ W,    // [64,128] row-major (B[k][c] = W[c][k])
    const float* __restrict__ bias, // [64]
    float* __restrict__ out,        // [n,64]
    int n) {
    const int lane = threadIdx.x & 31;
    const int wave = threadIdx.x >> 5;            // 0..7
    const int rowBase = (blockIdx.x * 8 + wave) * 16;

    const int mn = lane & 15;                     // row of A / col of B-D
    const int kh = lane >> 4;                     // K half-select {0,1}<->{2,3}

    // Clamp A row so EXEC stays all-ones for WMMA; stores are guarded below.
    int arow = rowBase + mn;
    if (arow >= n) arow = n - 1;
    const float* aptr = X + (size_t)arow * 128 + 2 * kh;
    const float* bptr = W + (size_t)mn * 128 + 2 * kh;  // col tile c: +c*16*128

    v8f acc0 = {}, acc1 = {}, acc2 = {}, acc3 = {};
#pragma unroll
    for (int k = 0; k < 128; k += 4) {
        v2f a, b0, b1, b2, b3;
        a.x  = aptr[k];                a.y  = aptr[k + 1];
        b0.x = bptr[k];                b0.y = bptr[k + 1];
        b1.x = bptr[16 * 128 + k];     b1.y = bptr[16 * 128 + k + 1];
        b2.x = bptr[32 * 128 + k];     b2.y = bptr[32 * 128 + k + 1];
        b3.x = bptr[48 * 128 + k];     b3.y = bptr[48 * 128 + k + 1];
        acc0 = __builtin_amdgcn_wmma_f32_16x16x4_f32(false, a, false, b0,
                                                     (short)0, acc0, false, false);
        acc1 = __builtin_amdgcn_wmma_f32_16x16x4_f32(false, a, false, b1,
                                                     (short)0, acc1, false, false);
        acc2 = __builtin_amdgcn_wmma_f32_16x16x4_f32(false, a, false, b2,
                                                     (short)0, acc2, false, false);
        acc3 = __builtin_amdgcn_wmma_f32_16x16x4_f32(false, a, false, b3,
                                                     (short)0, acc3, false, false);
    }

    // C/D layout: VGPR r holds row (r + 8*kh), col = mn within each 16-col tile.
    const float bv0 = bias[ 0 + mn];
    const float bv1 = bias[16 + mn];
    const float bv2 = bias[32 + mn];
    const float bv3 = bias[48 + mn];
#pragma unroll
    for (int r = 0; r < 8; ++r) {
        int orow = rowBase + r + kh * 8;
        if (orow < n) {
            float* o = out + (size_t)orow * 64 + mn;
            o[ 0] = acc0[r] + bv0;
            o[16] = acc1[r] + bv1;
            o[32] = acc2[r] + bv2;
            o[48] = acc3[r] + bv3;
        }
    }
}

// ---------------------------------------------------------------------------

extern "C" void kernel_launch(void* const* d_in, const int* in_sizes, int n_in,
                              void* d_out, int out_size, void* d_ws, size_t ws_size,
                              hipStream_t stream) {
    const float* x    = (const float*)d_in[0];   // [N,128]
    const int*   ei   = (const int*)d_in[1];     // [2,E]
    const float* W    = (const float*)d_in[2];   // [64,128]
    const float* bias = (const float*)d_in[3];   // [64]
    float*       out  = (float*)d_out;           // [N,64]

    const int N_ = in_sizes[0] / 128;
    const int E_ = in_sizes[1] / 2;
    const int* row = ei;        // source nodes
    const int* col = ei + E_;   // target nodes

    float* ws   = (float*)d_ws;
    float* dinv = ws;                                   // N floats (deg -> rsqrt in place)
    float* buf0 = ws + (size_t)((N_ + 63) / 64) * 64;   // N*128 floats
    float* buf1 = buf0 + (size_t)N_ * 128;              // N*128 floats

    const int T = 256;
    const int gN  = (N_ + T - 1) / T;
    const int gE  = (E_ + T - 1) / T;
    const int gN32 = (int)(((long long)N_ * 32 + T - 1) / T);
    const int gE32 = (int)(((long long)E_ * 32 + T - 1) / T);

    // degree norms
    sgc_init_deg<<<gN, T, 0, stream>>>(dinv, N_);
    sgc_count_deg<<<gE, T, 0, stream>>>(col, dinv, E_);
    sgc_rsqrt<<<gN, T, 0, stream>>>(dinv, N_);

    // hop 1: buf0 = D^-1/2 (A+I) D^-1/2 * x
    sgc_self<<<gN32, T, 0, stream>>>(x, dinv, buf0, N_);
    sgc_edge<<<gE32, T, 0, stream>>>(x, row, col, dinv, buf0, E_);

    // hop 2: buf1 = D^-1/2 (A+I) D^-1/2 * buf0
    sgc_self<<<gN32, T, 0, stream>>>(buf0, dinv, buf1, N_);
    sgc_edge<<<gE32, T, 0, stream>>>(buf0, row, col, dinv, buf1, E_);

    // out = buf1 @ W^T + bias   (f32 WMMA, 16x64 strip per wave)
    sgc_gemm_wmma<<<(N_ + 127) / 128, T, 0, stream>>>(buf1, W, bias, out, N_);
}